// GaussRenderer_63402307224125
// MI455X (gfx1250) — compile-verified
//
#include <hip/hip_runtime.h>
#include <math.h>
#include <stdint.h>

// ---------------- problem constants ----------------
#define NG      512
#define IMG_H   256
#define IMG_W   256
#define TILE_PX 25
#define REC     16          // floats per packed gaussian record

#define SH_C0   0.28209479177387814f
#define SH_C1   0.4886025119029199f
#define SH_C2_0 1.0925484305920792f
#define SH_C2_1 (-1.0925484305920792f)
#define SH_C2_2 0.31539156525252005f
#define SH_C2_3 (-1.0925484305920792f)
#define SH_C2_4 0.5462742152960396f
#define SH_C3_0 (-0.5900435899266435f)
#define SH_C3_1 2.890611442640554f
#define SH_C3_2 (-0.4570457994644658f)
#define SH_C3_3 0.3731763325901154f
#define SH_C3_4 (-0.4570457994644658f)
#define SH_C3_5 1.445305721320277f
#define SH_C3_6 (-0.5900435899266435f)

#define TANHALF 0.5773502691896257f            // tan(pi/6)
#define FOCALC  221.70250336881628f            // 0.5*256/tan(pi/6)

typedef __attribute__((ext_vector_type(2))) float v2f;
typedef __attribute__((ext_vector_type(8))) float v8f;

// record layout: 0 mx, 1 my, 2 conA, 3 conB, 4 conC, 5 op,
//                6 rminx, 7 rminy, 8 rmaxx, 9 rmaxy,
//                10 r, 11 g, 12 b, 13 depth, 14/15 pad

// ---------------- kernel 1: per-gaussian preprocess ----------------
__global__ void gs_preprocess(const float* __restrict__ means3D,
                              const float* __restrict__ scales,
                              const float* __restrict__ rotations,
                              const float* __restrict__ opacity,
                              const float* __restrict__ shs,
                              const float* __restrict__ vm,
                              const float* __restrict__ pm,
                              const float* __restrict__ cam,
                              float* __restrict__ pre,
                              float* __restrict__ out_mask,
                              float* __restrict__ out_radii)
{
  int i = blockIdx.x * blockDim.x + threadIdx.x;
  if (i >= NG) return;

  float mx3 = means3D[3*i+0], my3 = means3D[3*i+1], mz3 = means3D[3*i+2];

  // p_view = [m,1] @ viewmatrix  (row-vector convention)
  float pv0 = mx3*vm[0] + my3*vm[4] + mz3*vm[8]  + vm[12];
  float pv1 = mx3*vm[1] + my3*vm[5] + mz3*vm[9]  + vm[13];
  float pv2 = mx3*vm[2] + my3*vm[6] + mz3*vm[10] + vm[14];
  float pv3 = mx3*vm[3] + my3*vm[7] + mz3*vm[11] + vm[15];
  // p_h = p_view @ projmatrix
  float ph0 = pv0*pm[0] + pv1*pm[4] + pv2*pm[8]  + pv3*pm[12];
  float ph1 = pv0*pm[1] + pv1*pm[5] + pv2*pm[9]  + pv3*pm[13];
  float ph3 = pv0*pm[3] + pv1*pm[7] + pv2*pm[11] + pv3*pm[15];
  float inv = 1.0f / (ph3 + 1e-6f);
  float ppx = ph0*inv, ppy = ph1*inv;
  float depth = pv2;

  // ray direction for SH
  float rdx = mx3-cam[0], rdy = my3-cam[1], rdz = mz3-cam[2];
  float rn = rsqrtf(rdx*rdx + rdy*rdy + rdz*rdz);
  float x = rdx*rn, y = rdy*rn, z = rdz*rn;
  float xx=x*x, yy=y*y, zz=z*z, xy=x*y, yz=y*z, xz=x*z;

  float basis[16];
  basis[0]  = SH_C0;
  basis[1]  = -SH_C1*y;  basis[2] = SH_C1*z;  basis[3] = -SH_C1*x;
  basis[4]  = SH_C2_0*xy; basis[5] = SH_C2_1*yz;
  basis[6]  = SH_C2_2*(2.0f*zz-xx-yy);
  basis[7]  = SH_C2_3*xz; basis[8] = SH_C2_4*(xx-yy);
  basis[9]  = SH_C3_0*y*(3.0f*xx-yy);
  basis[10] = SH_C3_1*xy*z;
  basis[11] = SH_C3_2*y*(4.0f*zz-xx-yy);
  basis[12] = SH_C3_3*z*(2.0f*zz-3.0f*xx-3.0f*yy);
  basis[13] = SH_C3_4*x*(4.0f*zz-xx-yy);
  basis[14] = SH_C3_5*z*(xx-yy);
  basis[15] = SH_C3_6*x*(xx-3.0f*yy);

  float col[3];
#pragma unroll
  for (int c = 0; c < 3; ++c) {
    float s = 0.0f;
#pragma unroll
    for (int k = 0; k < 16; ++k) s += basis[k] * shs[(i*16 + k)*3 + c];
    col[c] = fmaxf(s + 0.5f, 0.0f);
  }

  // quaternion -> rotation
  float qr = rotations[4*i+0], qx = rotations[4*i+1];
  float qy = rotations[4*i+2], qz = rotations[4*i+3];
  float qn = rsqrtf(qr*qr + qx*qx + qy*qy + qz*qz);
  qr*=qn; qx*=qn; qy*=qn; qz*=qn;
  float R00=1.0f-2.0f*(qy*qy+qz*qz), R01=2.0f*(qx*qy-qr*qz), R02=2.0f*(qx*qz+qr*qy);
  float R10=2.0f*(qx*qy+qr*qz), R11=1.0f-2.0f*(qx*qx+qz*qz), R12=2.0f*(qy*qz-qr*qx);
  float R20=2.0f*(qx*qz-qr*qy), R21=2.0f*(qy*qz+qr*qx), R22=1.0f-2.0f*(qx*qx+qy*qy);

  float s0 = scales[3*i+0], s1 = scales[3*i+1], s2 = scales[3*i+2];
  float L00=R00*s0, L01=R01*s1, L02=R02*s2;
  float L10=R10*s0, L11=R11*s1, L12=R12*s2;
  float L20=R20*s0, L21=R21*s1, L22=R22*s2;

  float c00=L00*L00+L01*L01+L02*L02;
  float c01=L00*L10+L01*L11+L02*L12;
  float c02=L00*L20+L01*L21+L02*L22;
  float c11=L10*L10+L11*L11+L12*L12;
  float c12=L10*L20+L11*L21+L12*L22;
  float c22=L20*L20+L21*L21+L22*L22;

  // EWA splat Jacobian (t == p_view[:3])
  float tz = pv2;
  float invtz = 1.0f / tz;
  float lim = 1.3f * TANHALF;
  float txc = fminf(fmaxf(pv0*invtz, -lim), lim) * tz;
  float tyc = fminf(fmaxf(pv1*invtz, -lim), lim) * tz;
  float j00 = FOCALC*invtz;
  float j02 = -FOCALC*txc*invtz*invtz;
  float j11 = FOCALC*invtz;
  float j12 = -FOCALC*tyc*invtz*invtz;

  // M[a][c] = sum_b J[a][b] * Wr[c][b],  Wr[c][b] = vm[c*4+b]
  float M00 = j00*vm[0]  + j02*vm[2];
  float M01 = j00*vm[4]  + j02*vm[6];
  float M02 = j00*vm[8]  + j02*vm[10];
  float M10 = j11*vm[1]  + j12*vm[2];
  float M11 = j11*vm[5]  + j12*vm[6];
  float M12 = j11*vm[9]  + j12*vm[10];

  float t0 = c00*M00 + c01*M01 + c02*M02;
  float t1 = c01*M00 + c11*M01 + c12*M02;
  float t2 = c02*M00 + c12*M01 + c22*M02;
  float u0 = c00*M10 + c01*M11 + c02*M12;
  float u1 = c01*M10 + c11*M11 + c12*M12;
  float u2 = c02*M10 + c12*M11 + c22*M12;
  float cva = M00*t0 + M01*t1 + M02*t2 + 0.3f;
  float cvb = M10*t0 + M11*t1 + M12*t2;
  float cvc = M10*u0 + M11*u1 + M12*u2 + 0.3f;

  float mx = ((ppx + 1.0f)*(float)IMG_W - 1.0f)*0.5f;
  float my = ((ppy + 1.0f)*(float)IMG_H - 1.0f)*0.5f;

  float det = cva*cvc - cvb*cvb;
  float mid = 0.5f*(cva + cvc);
  float sq  = sqrtf(fmaxf(mid*mid - det, 0.1f));
  float lam = fmaxf(mid + sq, mid - sq);
  float radius = 3.0f * ceilf(sqrtf(lam));

  float rminx = fminf(fmaxf(mx - radius, 0.0f), (float)IMG_W - 1.0f);
  float rminy = fminf(fmaxf(my - radius, 0.0f), (float)IMG_H - 1.0f);
  float rmaxx = fminf(fmaxf(mx + radius, 0.0f), (float)IMG_W - 1.0f);
  float rmaxy = fminf(fmaxf(my + radius, 0.0f), (float)IMG_H - 1.0f);

  float icd  = 1.0f / det;
  float conA = cvc*icd, conB = -cvb*icd, conC = cva*icd;

  float* r = pre + i*REC;
  r[0]=mx;    r[1]=my;    r[2]=conA;  r[3]=conB;
  r[4]=conC;  r[5]=opacity[i]; r[6]=rminx; r[7]=rminy;
  r[8]=rmaxx; r[9]=rmaxy; r[10]=col[0]; r[11]=col[1];
  r[12]=col[2]; r[13]=depth; r[14]=0.0f; r[15]=0.0f;

  out_mask[i]  = (radius > 0.0f) ? 1.0f : 0.0f;
  out_radii[i] = radius;
}

// ---------------- kernel 2: stable rank-sort by depth ----------------
__global__ void __launch_bounds__(NG) gs_sort(const float* __restrict__ pre,
                                              float* __restrict__ sorted)
{
  __shared__ float sd[NG];
  int i = threadIdx.x;
  sd[i] = pre[i*REC + 13];
  __syncthreads();
  float di = sd[i];
  int rank = 0;
#pragma unroll 8
  for (int j = 0; j < NG; ++j) {
    float dj = sd[j];
    rank += (dj < di) || ((dj == di) && (j < i)) ? 1 : 0;
  }
  float* dst = sorted + rank*REC;
  const float* src = pre + i*REC;
#pragma unroll
  for (int f = 0; f < REC; ++f) dst[f] = src[f];
}

// ---------------- kernel 3: tiled render with fp32 WMMA accumulation --------
// Block = 256 threads = 8 waves; wave -> 16 pixels of one row.
// Gaussian table staged into LDS with GLOBAL_LOAD_ASYNC_TO_LDS_B128
// (ASYNCcnt path, no VGPR round-trip), then per chunk of 4 sorted gaussians
// one V_WMMA_F32_16X16X4_F32 accumulates
// [16 px x 4 gauss] x [4 gauss x (r,g,b,1,0..)] into the 16x16 f32 D tile.
__global__ void __launch_bounds__(256) gs_render(const float* __restrict__ sorted,
                                                 float* __restrict__ render_out,
                                                 float* __restrict__ acc_out)
{
  __shared__ float gtab[NG*REC];      // 32 KB gaussian table (LDS-resident)
  __shared__ float stage[8][64];      // per-wave D-tile staging (cols 0..3)

  __builtin_prefetch(sorted, 0, 0);   // global_prefetch_b8

  // ---- async global -> LDS staging of the sorted gaussian table ----
  {
    // LDS aperture: flat addr low 32 bits == LDS byte offset
    uint32_t ldsBase = (uint32_t)(uintptr_t)(&gtab[0]);
#pragma unroll
    for (int it = 0; it < (NG*REC*4) / (256*16); ++it) {      // 8 x b128/thread
      uint32_t byteOff = ((uint32_t)threadIdx.x + (uint32_t)it*256u) * 16u;
      uint32_t ldsAddr = ldsBase + byteOff;
      asm volatile("global_load_async_to_lds_b128 %0, %1, %2"
                   :
                   : "v"(ldsAddr), "v"(byteOff), "s"(sorted)
                   : "memory");
    }
    asm volatile("s_wait_asynccnt 0x0" ::: "memory");
  }

  // Inert wave-cluster path: plain launches have ClusterID==0, so the
  // cluster barrier is never executed (and would be a NOP per ISA anyway).
  if (__builtin_amdgcn_cluster_id_x() != 0) {
    __builtin_amdgcn_s_cluster_barrier();
  }

  __syncthreads();

  const int lane    = threadIdx.x & 31;
  const int wave    = threadIdx.x >> 5;
  const int M       = lane & 15;      // pixel row of A / column of B
  const int halfSel = lane >> 4;      // 0: K0,K1   1: K2,K3

  const int px = blockIdx.x*128 + wave*16 + M;
  const int py = blockIdx.y;
  const float fx = (float)px, fy = (float)py;
  const float tw = (float)((px / TILE_PX) * TILE_PX);
  const float th = (float)((py / TILE_PX) * TILE_PX);
  const float twmax = tw + (float)(TILE_PX - 1);
  const float thmax = th + (float)(TILE_PX - 1);

  float T = 1.0f;
  v8f acc = {0.0f,0.0f,0.0f,0.0f,0.0f,0.0f,0.0f,0.0f};
  const int cidx = (M < 3) ? M : 0;

  for (int k = 0; k < NG; k += 4) {
    float wj[4];
#pragma unroll
    for (int j = 0; j < 4; ++j) {
      const int base = (k + j) * REC;
      float4 q0 = *(const float4*)&gtab[base];       // mx,my,conA,conB
      float4 q1 = *(const float4*)&gtab[base + 4];   // conC,op,rminx,rminy
      float2 q2 = *(const float2*)&gtab[base + 8];   // rmaxx,rmaxy
      float otlx = fmaxf(q1.z, tw);
      float otly = fmaxf(q1.w, th);
      float obrx = fminf(q2.x, twmax);
      float obry = fminf(q2.y, thmax);
      float dx = fx - q0.x;
      float dy = fy - q0.y;
      float power = -0.5f*(q0.z*dx*dx + q1.x*dy*dy) - q0.w*dx*dy;
      float gv = __expf(power);
      float alpha = fminf(gv * q1.y, 0.99f);
      bool in = (obrx > otlx) && (obry > otly);
      alpha = in ? alpha : 0.0f;
      wj[j] = alpha * T;
      T *= (1.0f - alpha);
    }
    // A fragment: 16x4 fp32  (lanes 0-15: K0/K1, lanes 16-31: K2/K3)
    v2f a;
    a.x = halfSel ? wj[2] : wj[0];
    a.y = halfSel ? wj[3] : wj[1];
    // B fragment: 4x16 fp32, VGPR v holds rows K=v (lo lanes) / K=v+2 (hi lanes)
    int g0 = k + (halfSel ? 2 : 0);
    float c0 = gtab[g0*REC + 10 + cidx];
    float c1 = gtab[(g0 + 1)*REC + 10 + cidx];
    float bx = (M < 3) ? c0 : ((M == 3) ? 1.0f : 0.0f);
    float by = (M < 3) ? c1 : ((M == 3) ? 1.0f : 0.0f);
    v2f b; b.x = bx; b.y = by;

    acc = __builtin_amdgcn_wmma_f32_16x16x4_f32(false, a, false, b,
                                                (short)0, acc, false, false);
  }

  // D layout: VGPR v -> row v (lanes 0-15) / row v+8 (lanes 16-31), col = lane&15
  if (M < 4) {
#pragma unroll
    for (int v = 0; v < 8; ++v) {
      int row = v + (halfSel ? 8 : 0);
      stage[wave][row*4 + M] = acc[v];
    }
  }
  __syncthreads();

  if (lane < 16) {
    int row = lane;
    float r  = stage[wave][row*4 + 0];
    float g  = stage[wave][row*4 + 1];
    float bl = stage[wave][row*4 + 2];
    float av = stage[wave][row*4 + 3];
    float bg = 1.0f - av;                        // WHITE_BKGD
    int pix = py*IMG_W + (blockIdx.x*128 + wave*16 + row);
    render_out[pix*3 + 0] = r  + bg;
    render_out[pix*3 + 1] = g  + bg;
    render_out[pix*3 + 2] = bl + bg;
    acc_out[pix] = av;
  }
}

// ---------------- host launcher ----------------
extern "C" void kernel_launch(void* const* d_in, const int* in_sizes, int n_in,
                              void* d_out, int out_size, void* d_ws, size_t ws_size,
                              hipStream_t stream) {
  const float* means3D   = (const float*)d_in[0];
  const float* scales    = (const float*)d_in[1];
  const float* rotations = (const float*)d_in[2];
  const float* opacity   = (const float*)d_in[3];
  const float* shs       = (const float*)d_in[4];
  const float* viewm     = (const float*)d_in[5];
  const float* projm     = (const float*)d_in[6];
  const float* camc      = (const float*)d_in[7];

  float* out = (float*)d_out;
  float* render   = out;                                  // 256*256*3
  float* accalpha = out + IMG_H*IMG_W*3;                  // 256*256
  float* maskout  = out + IMG_H*IMG_W*3 + IMG_H*IMG_W;    // 512
  float* radiiout = maskout + NG;                         // 512

  float* pre    = (float*)d_ws;          // NG*REC floats
  float* sorted = pre + NG*REC;          // NG*REC floats

  gs_preprocess<<<dim3(4), dim3(128), 0, stream>>>(
      means3D, scales, rotations, opacity, shs, viewm, projm, camc,
      pre, maskout, radiiout);

  gs_sort<<<dim3(1), dim3(NG), 0, stream>>>(pre, sorted);

  gs_render<<<dim3(IMG_W/128, IMG_H), dim3(256), 0, stream>>>(
      sorted, render, accalpha);
}